// ThresholdedMSELoss_65936337928903
// MI455X (gfx1250) — compile-verified
//
#include <hip/hip_runtime.h>

typedef float v2f __attribute__((ext_vector_type(2)));
typedef float v8f __attribute__((ext_vector_type(8)));

__device__ __forceinline__ float wave_sum(float v) {
#pragma unroll
  for (int off = 16; off > 0; off >>= 1)
    v += __shfl_xor(v, off, 32);
  return v;
}

// Pass 1: grid-stride streaming; per-lane math feeds V_WMMA_F32_16X16X4_F32
// with an all-ones A so the f32 accumulator collects column sums (every row
// identical). Deterministic per-block partials -> d_ws.
__global__ void __launch_bounds__(256)
tmse_partial(const float4* __restrict__ pred4, const float4* __restrict__ lab4,
             float* __restrict__ ws, int total4, int n) {
  const v2f ones = {1.0f, 1.0f};
  v8f csq = {};  // squared-error accumulator (16x16 f32, rows replicated)
  v8f cac = {};  // accuracy-count accumulator
  const int stride = gridDim.x * blockDim.x;
  for (int i = blockIdx.x * blockDim.x + threadIdx.x; i < total4; i += stride) {
    float4 p = pred4[i];
    float4 l = lab4[i];
    __builtin_prefetch(pred4 + i + stride, 0, 0);  // global_prefetch_b8
    __builtin_prefetch(lab4 + i + stride, 0, 0);
    float e0 = p.x - l.x, e1 = p.y - l.y, e2 = p.z - l.z, e3 = p.w - l.w;
    // labels > 0, so (e/l <= 0.1) == (e <= 0.1*l): no division needed
    float f0 = (e0 <= 0.1f * l.x) ? 1.0f : 0.0f;
    float f1 = (e1 <= 0.1f * l.y) ? 1.0f : 0.0f;
    float f2 = (e2 <= 0.1f * l.z) ? 1.0f : 0.0f;
    float f3 = (e3 <= 0.1f * l.w) ? 1.0f : 0.0f;
    v2f bsq = { __builtin_fmaf(e0, e0, e1 * e1),
                __builtin_fmaf(e2, e2, e3 * e3) };
    v2f bac = { f0 + f1, f2 + f3 };
    // D = ones(16x4) x B(4x16) + C  -> accumulate sum of all B elements
    csq = __builtin_amdgcn_wmma_f32_16x16x4_f32(false, ones, false, bsq,
                                                (short)0, csq, false, false);
    cac = __builtin_amdgcn_wmma_f32_16x16x4_f32(false, ones, false, bac,
                                                (short)0, cac, false, false);
  }
  // C VGPR0 holds each of the 16 column sums exactly twice across 32 lanes.
  float ssq = wave_sum(csq[0]) * 0.5f;
  float sac = wave_sum(cac[0]) * 0.5f;

  __shared__ float smem[16];  // up to 8 waves * 2 values
  const int lane = threadIdx.x & 31;
  const int wave = threadIdx.x >> 5;
  if (lane == 0) { smem[2 * wave] = ssq; smem[2 * wave + 1] = sac; }
  __syncthreads();
  if (threadIdx.x == 0) {
    float t0 = 0.f, t1 = 0.f;
    const int nw = blockDim.x >> 5;
    for (int w = 0; w < nw; ++w) { t0 += smem[2 * w]; t1 += smem[2 * w + 1]; }
    if (blockIdx.x == 0) {  // scalar tail (n not multiple of 4); empty for N=2^24
      const float* pred = (const float*)pred4;
      const float* lab  = (const float*)lab4;
      for (int k = total4 * 4; k < n; ++k) {
        float e = pred[k] - lab[k];
        t0 += e * e;
        t1 += (e <= 0.1f * lab[k]) ? 1.0f : 0.0f;
      }
    }
    ws[2 * blockIdx.x]     = t0;
    ws[2 * blockIdx.x + 1] = t1;
  }
}

// Pass 2: single block reduces the per-block partials in fixed order.
__global__ void __launch_bounds__(256)
tmse_final(const float* __restrict__ ws, float* __restrict__ out,
           int nblocks, float invN) {
  float t0 = 0.f, t1 = 0.f;
  for (int i = threadIdx.x; i < nblocks; i += blockDim.x) {
    t0 += ws[2 * i];
    t1 += ws[2 * i + 1];
  }
  t0 = wave_sum(t0);
  t1 = wave_sum(t1);
  __shared__ float s0[8], s1[8];
  const int lane = threadIdx.x & 31;
  const int wave = threadIdx.x >> 5;
  if (lane == 0) { s0[wave] = t0; s1[wave] = t1; }
  __syncthreads();
  if (threadIdx.x == 0) {
    float a = 0.f, b = 0.f;
    for (int w = 0; w < 8; ++w) { a += s0[w]; b += s1[w]; }
    out[0] = a * invN;  // loss = mean(squared error)
    out[1] = b * invN;  // accuracy = mean(error/label <= 0.1)
  }
}

extern "C" void kernel_launch(void* const* d_in, const int* in_sizes, int n_in,
                              void* d_out, int out_size, void* d_ws, size_t ws_size,
                              hipStream_t stream) {
  const float4* p = (const float4*)d_in[0];
  const float4* l = (const float4*)d_in[1];
  const int n = in_sizes[0];
  const int total4 = n >> 2;

  int blocks = 2048;                                   // 16Ki wave32 waves in flight
  const int maxb = (int)(ws_size / (2 * sizeof(float)));
  if (blocks > maxb) blocks = maxb;
  if (blocks < 1) blocks = 1;

  tmse_partial<<<blocks, 256, 0, stream>>>(p, l, (float*)d_ws, total4, n);
  tmse_final<<<1, 256, 0, stream>>>((const float*)d_ws, (float*)d_out,
                                    blocks, 1.0f / (float)n);
}